// Self_Attn_54692113547349
// MI455X (gfx1250) — compile-verified
//
#include <hip/hip_runtime.h>
#include <hip/hip_bf16.h>
#include <math.h>

#define Hdim 2048
#define Bdim 32

typedef __attribute__((ext_vector_type(2))) float v2f;
typedef __attribute__((ext_vector_type(8))) float v8f;

// ---------------------------------------------------------------------------
// Kernel 1: q/k/v = x @ W^T + b  via V_WMMA_F32_16X16X4_F32
//   blockDim = 64 (2 waves). Each wave computes a 32x16 output tile
//   (two 16x16 accumulators stacked in M, sharing the B operand).
//   blockIdx.x : N-tile (32 cols per block), blockIdx.y : which matrix (q/k/v)
// ---------------------------------------------------------------------------
__global__ __launch_bounds__(64) void qkv_gemm_wmma(
    const float* __restrict__ x,
    const float* __restrict__ Wq, const float* __restrict__ bq,
    const float* __restrict__ Wk, const float* __restrict__ bk,
    const float* __restrict__ Wv, const float* __restrict__ bv,
    float* __restrict__ qkv)
{
    const int mat = blockIdx.y;                       // 0=q, 1=k, 2=v
    const float* __restrict__ W    = (mat == 0) ? Wq : (mat == 1) ? Wk : Wv;
    const float* __restrict__ bias = (mat == 0) ? bq : (mat == 1) ? bk : bv;
    float* __restrict__ out = qkv + (size_t)mat * (Bdim * Hdim);

    const int wave  = threadIdx.x >> 5;               // 0..1
    const int lane  = threadIdx.x & 31;
    const int half  = lane >> 4;                      // 0: K={0,1}, 1: K={2,3}
    const int l16   = lane & 15;
    const int nBase = blockIdx.x * 32 + wave * 16;

    // A layout (16x4 f32): lanes 0-15 -> M=0..15 K={k,k+1}; lanes 16-31 -> K={k+2,k+3}
    const float* __restrict__ xa_lo = x + (size_t)l16 * Hdim;          // M rows 0..15
    const float* __restrict__ xa_hi = x + (size_t)(16 + l16) * Hdim;   // M rows 16..31
    // B layout (4x16 f32): lane half picks K pair, l16 picks column n.
    // q = x @ W^T  =>  B[kk][n] = W[n][kk]  (row n of W is contiguous in K: ideal streaming)
    const float* __restrict__ wb = W + (size_t)(nBase + l16) * Hdim;
    const int kofs = half * 2;

    v8f c_lo = {};   // M = 0..15
    v8f c_hi = {};   // M = 16..31

    #pragma unroll 4
    for (int k = 0; k < Hdim; k += 4) {
        v2f a_lo = *(const v2f*)(xa_lo + k + kofs);
        v2f a_hi = *(const v2f*)(xa_hi + k + kofs);
        v2f b    = *(const v2f*)(wb    + k + kofs);
        c_lo = __builtin_amdgcn_wmma_f32_16x16x4_f32(
            false, a_lo, false, b, (short)0, c_lo, false, false);
        c_hi = __builtin_amdgcn_wmma_f32_16x16x4_f32(
            false, a_hi, false, b, (short)0, c_hi, false, false);
    }

    // C/D layout: VGPR r, lanes 0-15 -> (M=r, N=l16), lanes 16-31 -> (M=r+8, N=l16)
    const int n  = nBase + l16;
    const float bb = bias[n];
    #pragma unroll
    for (int r = 0; r < 8; ++r) {
        const int m = r + half * 8;
        out[(size_t)m        * Hdim + n] = c_lo[r] + bb;
        out[(size_t)(m + 16) * Hdim + n] = c_hi[r] + bb;
    }
}

// ---------------------------------------------------------------------------
// Kernel 2: rank-1 softmax-attention, transcendental bound.
//   out[b,h] = gamma * (sum_j v_j * e^{q_h k_j - m}) / (sum_j e^{q_h k_j - m}) + x[b,h]
//   with exact row max m = q_h * (q_h >= 0 ? kmax : kmin).
//   grid = (32 batches, 8 h-chunks), block = 256 threads, one h per thread.
// ---------------------------------------------------------------------------
__global__ __launch_bounds__(256) void rank1_attn(
    const float* __restrict__ qkv,
    const float* __restrict__ x,
    const float* __restrict__ gamma,
    float* __restrict__ out)
{
    const int b = blockIdx.x;
    const float* __restrict__ q = qkv + (size_t)b * Hdim;
    const float* __restrict__ k = qkv + (size_t)(Bdim + b) * Hdim;
    const float* __restrict__ v = qkv + (size_t)(2 * Bdim + b) * Hdim;

    __shared__ float sk[Hdim];
    __shared__ float sv[Hdim];
    __shared__ float red_max[8];
    __shared__ float red_min[8];
    __shared__ float s_kmax, s_kmin;

    // Stage k, v into LDS with 128-bit loads
    for (int i = threadIdx.x; i < Hdim / 4; i += blockDim.x) {
        ((float4*)sk)[i] = ((const float4*)k)[i];
        ((float4*)sv)[i] = ((const float4*)v)[i];
    }
    __syncthreads();

    // Block-wide kmax / kmin (wave32 shuffle tree, then LDS across 8 waves)
    float kmax = -INFINITY, kmin = INFINITY;
    for (int j = threadIdx.x; j < Hdim; j += blockDim.x) {
        const float kj = sk[j];
        kmax = fmaxf(kmax, kj);
        kmin = fminf(kmin, kj);
    }
    #pragma unroll
    for (int o = 16; o > 0; o >>= 1) {
        kmax = fmaxf(kmax, __shfl_xor(kmax, o, 32));
        kmin = fminf(kmin, __shfl_xor(kmin, o, 32));
    }
    const int wave = threadIdx.x >> 5;
    const int lane = threadIdx.x & 31;
    if (lane == 0) { red_max[wave] = kmax; red_min[wave] = kmin; }
    __syncthreads();
    if (threadIdx.x == 0) {
        float mx = red_max[0], mn = red_min[0];
        #pragma unroll
        for (int w = 1; w < 8; ++w) {
            mx = fmaxf(mx, red_max[w]);
            mn = fminf(mn, red_min[w]);
        }
        s_kmax = mx; s_kmin = mn;
    }
    __syncthreads();

    const int h  = blockIdx.y * 256 + threadIdx.x;
    const float qh = q[h];
    const float m  = (qh >= 0.0f) ? qh * s_kmax : qh * s_kmin;

    float se = 0.0f, sve = 0.0f;
    #pragma unroll 2
    for (int j = 0; j < Hdim; j += 4) {
        const float4 k4 = *(const float4*)(sk + j);   // LDS broadcast
        const float4 v4 = *(const float4*)(sv + j);
        const float e0 = __expf(fmaf(qh, k4.x, -m));
        const float e1 = __expf(fmaf(qh, k4.y, -m));
        const float e2 = __expf(fmaf(qh, k4.z, -m));
        const float e3 = __expf(fmaf(qh, k4.w, -m));
        se  += (e0 + e1) + (e2 + e3);
        sve  = fmaf(v4.x, e0, sve);
        sve  = fmaf(v4.y, e1, sve);
        sve  = fmaf(v4.z, e2, sve);
        sve  = fmaf(v4.w, e3, sve);
    }

    const size_t idx = (size_t)b * Hdim + h;
    out[idx] = fmaf(gamma[0], sve / se, x[idx]);
}

// ---------------------------------------------------------------------------
extern "C" void kernel_launch(void* const* d_in, const int* in_sizes, int n_in,
                              void* d_out, int out_size, void* d_ws, size_t ws_size,
                              hipStream_t stream) {
    const float* x     = (const float*)d_in[0];
    const float* Wq    = (const float*)d_in[1];
    const float* bq    = (const float*)d_in[2];
    const float* Wk    = (const float*)d_in[3];
    const float* bk    = (const float*)d_in[4];
    const float* Wv    = (const float*)d_in[5];
    const float* bv    = (const float*)d_in[6];
    const float* gamma = (const float*)d_in[7];
    float* out = (float*)d_out;
    float* qkv = (float*)d_ws;   // 3 * 32 * 2048 floats = 768 KB, fully overwritten

    dim3 g1(Hdim / 32, 3);       // 64 N-tiles x {q,k,v} = 192 blocks
    qkv_gemm_wmma<<<g1, 64, 0, stream>>>(x, Wq, bq, Wk, bk, Wv, bv, qkv);

    dim3 g2(Bdim, Hdim / 256);   // 32 batches x 8 h-chunks = 256 blocks
    rank1_attn<<<g2, 256, 0, stream>>>(qkv, x, gamma, out);
}